// VecInt_41446434406565
// MI455X (gfx1250) — compile-verified
//
#include <hip/hip_runtime.h>

#define D_ 160
#define H_ 192
#define W_ 160
#define N_ (D_ * H_ * W_)

struct F3 { float x, y, z; };

template <bool FIRST>
__global__ __launch_bounds__(W_) void vecint_warp_step(
    const float* __restrict__ vin, float* __restrict__ vout)
{
    constexpr float s = FIRST ? (1.0f / 128.0f) : 1.0f;  // 1/2^NB_STEPS on step 1

    const int k = threadIdx.x;            // W index
    const int j = blockIdx.x;             // H index
    const int i = blockIdx.y;             // D index
    const int idx = (i * H_ + j) * W_ + k;

    const F3* __restrict__ vin3 = reinterpret_cast<const F3*>(vin);
    F3* __restrict__ vout3 = reinterpret_cast<F3*>(vout);

    // Speculative prefetch of the streaming flow read ~32KB ahead, WGP scope
    // (pull into all cache levels). OOB prefetches are silently dropped.
    __builtin_prefetch(reinterpret_cast<const char*>(vin3 + idx) + 32768, 0, 3);

    const F3 f = vin3[idx];               // global_load_b96 (coalesced)

    // Sample position = identity grid + s*flow, edge-clamped (s*f exact: s = 2^-7).
    const float px = fminf(fmaxf((float)i + s * f.x, 0.0f), (float)(D_ - 1));
    const float py = fminf(fmaxf((float)j + s * f.y, 0.0f), (float)(H_ - 1));
    const float pz = fminf(fmaxf((float)k + s * f.z, 0.0f), (float)(W_ - 1));

    const float fx = floorf(px), fy = floorf(py), fz = floorf(pz);
    const int i0 = (int)fx, j0 = (int)fy, k0 = (int)fz;
    const int i1 = min(i0 + 1, D_ - 1);
    const int j1 = min(j0 + 1, H_ - 1);
    const int k1 = min(k0 + 1, W_ - 1);

    const float wi = px - fx, wj = py - fy, wk = pz - fz;
    const float ui = 1.0f - wi, uj = 1.0f - wj, uk = 1.0f - wk;

    const int base = (i0 * H_ + j0) * W_ + k0;
    const int di = (i1 - i0) * (H_ * W_);
    const int dj = (j1 - j0) * W_;
    const int dk = (k1 - k0);

    // 8 corner gathers (spatially local -> L0/L2 hits; 192MB L2 holds the field)
    const F3 g000 = vin3[base];
    const F3 g001 = vin3[base + dk];
    const F3 g010 = vin3[base + dj];
    const F3 g011 = vin3[base + dj + dk];
    const F3 g100 = vin3[base + di];
    const F3 g101 = vin3[base + di + dk];
    const F3 g110 = vin3[base + di + dj];
    const F3 g111 = vin3[base + di + dj + dk];

    const float w000 = ui * uj * uk;
    const float w001 = ui * uj * wk;
    const float w010 = ui * wj * uk;
    const float w011 = ui * wj * wk;
    const float w100 = wi * uj * uk;
    const float w101 = wi * uj * wk;
    const float w110 = wi * wj * uk;
    const float w111 = wi * wj * wk;

    // Accumulate in the reference's loop order (i-major, then j, then k).
    float tx = w000 * g000.x;
    float ty = w000 * g000.y;
    float tz = w000 * g000.z;
    tx = fmaf(w001, g001.x, tx); ty = fmaf(w001, g001.y, ty); tz = fmaf(w001, g001.z, tz);
    tx = fmaf(w010, g010.x, tx); ty = fmaf(w010, g010.y, ty); tz = fmaf(w010, g010.z, tz);
    tx = fmaf(w011, g011.x, tx); ty = fmaf(w011, g011.y, ty); tz = fmaf(w011, g011.z, tz);
    tx = fmaf(w100, g100.x, tx); ty = fmaf(w100, g100.y, ty); tz = fmaf(w100, g100.z, tz);
    tx = fmaf(w101, g101.x, tx); ty = fmaf(w101, g101.y, ty); tz = fmaf(w101, g101.z, tz);
    tx = fmaf(w110, g110.x, tx); ty = fmaf(w110, g110.y, ty); tz = fmaf(w110, g110.z, tz);
    tx = fmaf(w111, g111.x, tx); ty = fmaf(w111, g111.y, ty); tz = fmaf(w111, g111.z, tz);

    // out = s*(v_raw + trilerp_raw) == v + warp(v, v) with v = s*v_raw
    F3 o;
    if (FIRST) {
        o.x = s * (f.x + tx);
        o.y = s * (f.y + ty);
        o.z = s * (f.z + tz);
    } else {
        o.x = f.x + tx;
        o.y = f.y + ty;
        o.z = f.z + tz;
    }
    vout3[idx] = o;                       // global_store_b96 (coalesced)
}

extern "C" void kernel_launch(void* const* d_in, const int* in_sizes, int n_in,
                              void* d_out, int out_size, void* d_ws, size_t ws_size,
                              hipStream_t stream) {
    const float* vec = (const float*)d_in[0];
    float* out = (float*)d_out;
    float* tmp = (float*)d_ws;            // one N*3 float ping-pong buffer

    const long per_batch = (long)N_ * 3;
    const int B = (int)(in_sizes[0] / per_batch);

    const dim3 grid(H_, D_);
    const dim3 block(W_);

    for (int b = 0; b < B; ++b) {
        const float* src = vec + (size_t)b * per_batch;
        float* dst = out + (size_t)b * per_batch;

        // step 1: v1 = v0 + warp(v0,v0), v0 = vec/128 (scale folded in)
        vecint_warp_step<true ><<<grid, block, 0, stream>>>(src, dst);
        // steps 2..7: ping-pong dst <-> tmp; odd steps land in dst
        vecint_warp_step<false><<<grid, block, 0, stream>>>(dst, tmp); // step 2
        vecint_warp_step<false><<<grid, block, 0, stream>>>(tmp, dst); // step 3
        vecint_warp_step<false><<<grid, block, 0, stream>>>(dst, tmp); // step 4
        vecint_warp_step<false><<<grid, block, 0, stream>>>(tmp, dst); // step 5
        vecint_warp_step<false><<<grid, block, 0, stream>>>(dst, tmp); // step 6
        vecint_warp_step<false><<<grid, block, 0, stream>>>(tmp, dst); // step 7 -> d_out
    }
}